// MHAttentionMap_54391465836797
// MI455X (gfx1250) — compile-verified
//
#include <hip/hip_runtime.h>

// ---------------------------------------------------------------------------
// MHAttentionMap (DETR-style) for MI455X / gfx1250, wave32, fp32 WMMA.
// B=4, NF=300, QUERY_DIM=512, HIDDEN=512, NUM_HEADS=8, HEAD_DIM=64, H=W=80.
// Wave tile: 32(M) x 64(N)  -> 2 A fragments, 4 B tiles, 8 f32 WMMAs / k-step.
// qp workspace padded to 320 rows so all stores/loads in the M-boundary
// region are unconditional (no divergent EXEC-masked store ladders).
// ---------------------------------------------------------------------------

#define BATCH     4
#define NF        300
#define NFP       320                     // padded query rows in qp workspace
#define QD        512
#define HID       512
#define NHEADS    8
#define HDIM      64
#define HW        6400                     // 80*80
#define PB        ((size_t)NF*NHEADS*HW)   // 15,360,000 elements per batch
#define NORMF     0.125f                   // (512/8)^-0.5

typedef float v2f __attribute__((ext_vector_type(2)));
typedef float v8f __attribute__((ext_vector_type(8)));

static __device__ __forceinline__ v8f wmma4(v2f a, v2f b, v8f c) {
  // D = A(16x4, f32) * B(4x16, f32) + C(16x16, f32)
  return __builtin_amdgcn_wmma_f32_16x16x4_f32(false, a, false, b, (short)0, c,
                                               false, false);
}
static __device__ __forceinline__ v2f ld2(const float* __restrict__ p) {
  return *(const v2f*)p;                   // 8B-aligned -> global_load_b64
}

// ---------------------------------------------------------------------------
// Kernel 1: q projection, scaled, into padded qp [B][NFP][HID].
//   qp[b, m, n] = NORMF * ( sum_k q[b,m,k] * q_w[n,k] + q_b[n] )
//   Block tile 128x64, wave tile 32x64; waves fully above row 320 exit.
// ---------------------------------------------------------------------------
__global__ __launch_bounds__(128) void qproj_kernel(
    const float* __restrict__ q, const float* __restrict__ qw,
    const float* __restrict__ qb, float* __restrict__ qp)
{
  const int b    = blockIdx.y;
  const int NTB  = HID / 64;                 // 8
  const int bm   = blockIdx.x / NTB;
  const int bn   = blockIdx.x % NTB;
  const int wave = threadIdx.x >> 5;
  const int m0   = bm * 128 + wave * 32;
  if (m0 >= NFP) return;                     // wave-uniform: EXEC stays full

  const int lane = threadIdx.x & 31;
  const int lm   = lane & 15;
  const int koff = (lane >> 4) << 1;         // 0 or 2
  const int n0   = bn * 64;

  const float* A = q + (size_t)b * NF * QD;  // A[m][k], lda=512
  // Clamp out-of-range input rows to row 0: their products land only in
  // padded qp rows (never consumed as real queries), so loads stay
  // unconditional and the values stay finite.
  const int r0 = m0 + lm,      r1 = m0 + 16 + lm;
  const size_t ab0 = (size_t)(r0 < NF ? r0 : 0) * QD;
  const size_t ab1 = (size_t)(r1 < NF ? r1 : 0) * QD;

  v8f acc[2][4];
  #pragma unroll
  for (int mi = 0; mi < 2; ++mi)
    #pragma unroll
    for (int t = 0; t < 4; ++t) acc[mi][t] = (v8f){};

  #pragma unroll 2
  for (int k = 0; k < QD; k += 4) {
    const v2f a0 = ld2(A + ab0 + k + koff);
    const v2f a1 = ld2(A + ab1 + k + koff);
    #pragma unroll
    for (int t = 0; t < 4; ++t) {
      const int n = n0 + t * 16 + lm;
      const v2f bb = ld2(qw + (size_t)n * QD + k + koff); // B[k][n]=q_w[n][k]
      acc[0][t] = wmma4(a0, bb, acc[0][t]);
      acc[1][t] = wmma4(a1, bb, acc[1][t]);
    }
  }

  const int jbase = (lane >> 4) << 3;        // 0 or 8
  float* C = qp + (size_t)b * NFP * HID;
  #pragma unroll
  for (int mi = 0; mi < 2; ++mi) {
    #pragma unroll
    for (int t = 0; t < 4; ++t) {
      const int n = n0 + t * 16 + lm;
      const float bias = qb[n];
      #pragma unroll
      for (int j = 0; j < 8; ++j) {
        const int r = m0 + mi * 16 + jbase + j;          // always < NFP
        C[(size_t)r * HID + n] = NORMF * (acc[mi][t][j] + bias);
      }
    }
  }
}

// ---------------------------------------------------------------------------
// Kernel 2: k projection (1x1 conv as channel GEMM).
//   kp[b, m, n] = sum_k k_w[m,k] * kin[b,k,n] + k_b[m]
//   M=512, N=6400, K=512 (exact tiles, no guards).  Block 128x64, wave 32x64.
// ---------------------------------------------------------------------------
__global__ __launch_bounds__(128) void kproj_kernel(
    const float* __restrict__ kw, const float* __restrict__ kin,
    const float* __restrict__ kb, float* __restrict__ kp)
{
  const int b    = blockIdx.y;
  const int NTB  = HW / 64;                  // 100
  const int bm   = blockIdx.x / NTB;
  const int bn   = blockIdx.x % NTB;
  const int wave = threadIdx.x >> 5;
  const int lane = threadIdx.x & 31;
  const int lm   = lane & 15;
  const int koff = (lane >> 4) << 1;
  const int m0   = bm * 128 + wave * 32;
  const int n0   = bn * 64;

  const float* B = kin + (size_t)b * QD * HW;     // B[k][n], ldb=6400
  const size_t ab0 = (size_t)(m0 + lm) * QD;      // k_w[m][k], lda=512
  const size_t ab1 = (size_t)(m0 + 16 + lm) * QD;

  v8f acc[2][4];
  #pragma unroll
  for (int mi = 0; mi < 2; ++mi)
    #pragma unroll
    for (int t = 0; t < 4; ++t) acc[mi][t] = (v8f){};

  #pragma unroll 2
  for (int k = 0; k < QD; k += 4) {
    const v2f a0 = ld2(kw + ab0 + k + koff);
    const v2f a1 = ld2(kw + ab1 + k + koff);
    #pragma unroll
    for (int t = 0; t < 4; ++t) {
      const int n = n0 + t * 16 + lm;
      v2f bb;                                     // rows k+koff, k+koff+1
      bb.x = B[(size_t)(k + koff)     * HW + n];
      bb.y = B[(size_t)(k + koff + 1) * HW + n];
      acc[0][t] = wmma4(a0, bb, acc[0][t]);
      acc[1][t] = wmma4(a1, bb, acc[1][t]);
    }
  }

  const int jbase = (lane >> 4) << 3;
  float* C = kp + (size_t)b * HID * HW;
  #pragma unroll
  for (int mi = 0; mi < 2; ++mi) {
    #pragma unroll
    for (int j = 0; j < 8; ++j) {
      const int r = m0 + mi * 16 + jbase + j;
      const float bias = kb[r];                   // per-row bias (channel e)
      #pragma unroll
      for (int t = 0; t < 4; ++t) {
        const int n = n0 + t * 16 + lm;
        C[(size_t)r * HW + n] = acc[mi][t][j] + bias;
      }
    }
  }
}

// ---------------------------------------------------------------------------
// Kernel 3: per-head QK scores, masked, written straight into d_out.
//   s[b, m, h, n] = sum_c qp[b,m,h*64+c] * kp[b,h*64+c,n];  masked -> -inf
//   M=300, N=6400, K=64 per (batch, head).  Reads padded qp (320 rows)
//   unconditionally; store epilogue uses a wave-uniform full-tile fast path.
// ---------------------------------------------------------------------------
__global__ __launch_bounds__(128) void scores_kernel(
    const float* __restrict__ qp, const float* __restrict__ kp,
    const int* __restrict__ mask, float* __restrict__ out)
{
  const int b    = blockIdx.y >> 3;
  const int h    = blockIdx.y & 7;
  const int NTB  = HW / 64;                  // 100
  const int bm   = blockIdx.x / NTB;
  const int bn   = blockIdx.x % NTB;
  const int wave = threadIdx.x >> 5;
  const int m0   = bm * 128 + wave * 32;
  if (m0 >= NF) return;                      // wave-uniform early exit

  const int lane = threadIdx.x & 31;
  const int lm   = lane & 15;
  const int koff = (lane >> 4) << 1;
  const int n0   = bn * 64;

  const float* A = qp + (size_t)b * NFP * HID + h * HDIM;             // lda=512
  const float* B = kp + (size_t)b * HID * HW + (size_t)h * HDIM * HW; // ldb=6400
  const size_t ab0 = (size_t)(m0 + lm) * HID;        // rows < 320: in padding
  const size_t ab1 = (size_t)(m0 + 16 + lm) * HID;

  v8f acc[2][4];
  #pragma unroll
  for (int mi = 0; mi < 2; ++mi)
    #pragma unroll
    for (int t = 0; t < 4; ++t) acc[mi][t] = (v8f){};

  #pragma unroll
  for (int k = 0; k < HDIM; k += 4) {
    const v2f a0 = ld2(A + ab0 + k + koff);
    const v2f a1 = ld2(A + ab1 + k + koff);
    #pragma unroll
    for (int t = 0; t < 4; ++t) {
      const int n = n0 + t * 16 + lm;
      v2f bb;
      bb.x = B[(size_t)(k + koff)     * HW + n];
      bb.y = B[(size_t)(k + koff + 1) * HW + n];
      acc[0][t] = wmma4(a0, bb, acc[0][t]);
      acc[1][t] = wmma4(a1, bb, acc[1][t]);
    }
  }

  const int jbase = (lane >> 4) << 3;
  const float NEG_INF = -__builtin_inff();
  #pragma unroll
  for (int mi = 0; mi < 2; ++mi) {
    const int tb = m0 + mi * 16;                 // wave-uniform tile base
    if (tb + 15 < NF) {
      // Fast path: whole 16-row tile in range -> unconditional stores.
      #pragma unroll
      for (int t = 0; t < 4; ++t) {
        const int n = n0 + t * 16 + lm;
        const bool masked = mask[(size_t)b * HW + n] != 0;
        #pragma unroll
        for (int j = 0; j < 8; ++j) {
          const int r = tb + jbase + j;
          const float v = masked ? NEG_INF : acc[mi][t][j];
          out[(((size_t)b * NF + r) * NHEADS + h) * HW + n] = v;
        }
      }
    } else if (tb < NF) {
      // Boundary tile (rows 288..303): guarded stores.
      #pragma unroll
      for (int t = 0; t < 4; ++t) {
        const int n = n0 + t * 16 + lm;
        const bool masked = mask[(size_t)b * HW + n] != 0;
        #pragma unroll
        for (int j = 0; j < 8; ++j) {
          const int r = tb + jbase + j;
          if (r < NF) {
            const float v = masked ? NEG_INF : acc[mi][t][j];
            out[(((size_t)b * NF + r) * NHEADS + h) * HW + n] = v;
          }
        }
      }
    }
  }
}

// ---------------------------------------------------------------------------
// Softmax over the full 15.36M-element flatten per batch (matches reference).
// Deterministic two-level reductions, float4-vectorized streaming passes.
// ---------------------------------------------------------------------------
static __device__ __forceinline__ float block_reduce_max(float v, float* s) {
  s[threadIdx.x] = v; __syncthreads();
  for (int o = blockDim.x >> 1; o > 0; o >>= 1) {
    if ((int)threadIdx.x < o) s[threadIdx.x] = fmaxf(s[threadIdx.x], s[threadIdx.x + o]);
    __syncthreads();
  }
  return s[0];
}
static __device__ __forceinline__ float block_reduce_sum(float v, float* s) {
  s[threadIdx.x] = v; __syncthreads();
  for (int o = blockDim.x >> 1; o > 0; o >>= 1) {
    if ((int)threadIdx.x < o) s[threadIdx.x] += s[threadIdx.x + o];
    __syncthreads();
  }
  return s[0];
}

__global__ void pmax_kernel(const float* __restrict__ x, float* __restrict__ part)
{
  const int b = blockIdx.y;
  const float4* p = (const float4*)(x + (size_t)b * PB);
  const size_t n4 = PB / 4;
  float m = -__builtin_inff();
  for (size_t i = blockIdx.x * (size_t)blockDim.x + threadIdx.x; i < n4;
       i += (size_t)gridDim.x * blockDim.x) {
    float4 v = p[i];
    m = fmaxf(m, fmaxf(fmaxf(v.x, v.y), fmaxf(v.z, v.w)));
  }
  __shared__ float s[256];
  m = block_reduce_max(m, s);
  if (threadIdx.x == 0) part[(size_t)b * gridDim.x + blockIdx.x] = m;
}

__global__ void rmax_final(const float* __restrict__ part, float* __restrict__ bmax, int nb)
{
  const int b = blockIdx.x;
  float m = -__builtin_inff();
  for (int i = threadIdx.x; i < nb; i += blockDim.x) m = fmaxf(m, part[(size_t)b * nb + i]);
  __shared__ float s[256];
  m = block_reduce_max(m, s);
  if (threadIdx.x == 0) bmax[b] = m;
}

__global__ void expsum_kernel(float* __restrict__ x, const float* __restrict__ bmax,
                              float* __restrict__ part)
{
  const int b = blockIdx.y;
  const float mx = bmax[b];
  float4* p = (float4*)(x + (size_t)b * PB);
  const size_t n4 = PB / 4;
  float acc = 0.0f;
  for (size_t i = blockIdx.x * (size_t)blockDim.x + threadIdx.x; i < n4;
       i += (size_t)gridDim.x * blockDim.x) {
    float4 v = p[i];
    v.x = __expf(v.x - mx); v.y = __expf(v.y - mx);
    v.z = __expf(v.z - mx); v.w = __expf(v.w - mx);
    p[i] = v;
    acc += (v.x + v.y) + (v.z + v.w);
  }
  __shared__ float s[256];
  acc = block_reduce_sum(acc, s);
  if (threadIdx.x == 0) part[(size_t)b * gridDim.x + blockIdx.x] = acc;
}

__global__ void rsum_final(const float* __restrict__ part, float* __restrict__ binv, int nb)
{
  const int b = blockIdx.x;
  float acc = 0.0f;
  for (int i = threadIdx.x; i < nb; i += blockDim.x) acc += part[(size_t)b * nb + i];
  __shared__ float s[256];
  acc = block_reduce_sum(acc, s);
  if (threadIdx.x == 0) binv[b] = 1.0f / acc;
}

__global__ void scale_kernel(float* __restrict__ x, const float* __restrict__ binv)
{
  const int b = blockIdx.y;
  const float sc = binv[b];
  float4* p = (float4*)(x + (size_t)b * PB);
  const size_t n4 = PB / 4;
  for (size_t i = blockIdx.x * (size_t)blockDim.x + threadIdx.x; i < n4;
       i += (size_t)gridDim.x * blockDim.x) {
    float4 v = p[i];
    v.x *= sc; v.y *= sc; v.z *= sc; v.w *= sc;
    p[i] = v;
  }
}

// ---------------------------------------------------------------------------
extern "C" void kernel_launch(void* const* d_in, const int* in_sizes, int n_in,
                              void* d_out, int out_size, void* d_ws, size_t ws_size,
                              hipStream_t stream) {
  const float* q    = (const float*)d_in[0];
  const float* kin  = (const float*)d_in[1];
  const int*   mask = (const int*)d_in[2];
  const float* qw   = (const float*)d_in[3];
  const float* qb   = (const float*)d_in[4];
  const float* kw   = (const float*)d_in[5];
  const float* kb   = (const float*)d_in[6];
  float* out = (float*)d_out;

  float* ws    = (float*)d_ws;
  float* qp    = ws;                              // 4*320*512    = 655,360
  float* kp    = qp + (size_t)BATCH * NFP * HID;  // 4*512*6400   = 13,107,200
  float* part1 = kp + (size_t)BATCH * HID * HW;   // 4*512
  float* part2 = part1 + BATCH * 512;             // 4*512
  float* bmax  = part2 + BATCH * 512;             // 4
  float* binv  = bmax + BATCH;                    // 4

  // GEMM phase (fp32 WMMA, wave tile 32x64)
  qproj_kernel <<<dim3(3 * 8,    BATCH),          128, 0, stream>>>(q,  qw, qb, qp);
  kproj_kernel <<<dim3(4 * 100,  BATCH),          128, 0, stream>>>(kw, kin, kb, kp);
  scores_kernel<<<dim3(3 * 100,  BATCH * NHEADS), 128, 0, stream>>>(qp, kp, mask, out);

  // Global masked softmax per batch
  pmax_kernel  <<<dim3(512, BATCH), 256, 0, stream>>>(out, part1);
  rmax_final   <<<BATCH,            256, 0, stream>>>(part1, bmax, 512);
  expsum_kernel<<<dim3(512, BATCH), 256, 0, stream>>>(out, bmax, part2);
  rsum_final   <<<BATCH,            256, 0, stream>>>(part2, binv, 512);
  scale_kernel <<<dim3(512, BATCH), 256, 0, stream>>>(out, binv);
}